// MambaClassifier_2534030705293
// MI455X (gfx1250) — compile-verified
//
#include <hip/hip_runtime.h>
#include <math.h>

// ---- problem constants (from reference) ----
#define D_MODEL 6
#define ED      48
#define NSTATE  32
#define DCONV   16
#define NLAYERS 2
#define NCLS    4
#define BATCH   16
#define SEQL    1024
#define BL      (BATCH * SEQL)       // 16384 tokens
#define DBC_PAD 80                   // 65 real cols (1 dt + 32 B + 32 C) padded to 5x16 tiles
#define WPP_N   (24 * 160)           // packed weight: 24 K-pairs x 80 cols x 2 = 3840 floats

typedef __attribute__((ext_vector_type(2))) float v2f;
typedef __attribute__((ext_vector_type(8))) float v8f;
typedef __attribute__((ext_vector_type(4))) unsigned int v4u;
typedef __attribute__((ext_vector_type(8))) int v8i;
typedef __attribute__((ext_vector_type(4))) int v4i;

// ---------------------------------------------------------------------------
// Kernel 1: fused rmsnorm + in_proj.  One thread per (token, output j in 0..95).
// ---------------------------------------------------------------------------
__global__ void token_pre_k(const float* __restrict__ hin,
                            const float* __restrict__ norm_w,
                            const float* __restrict__ in_w,
                            float* __restrict__ xs_raw,
                            float* __restrict__ zbuf) {
  int t = blockIdx.x * blockDim.x + threadIdx.x;
  if (t >= BL * (2 * ED)) return;
  int tok = t / (2 * ED);
  int j   = t % (2 * ED);
  const float* hrow = hin + tok * D_MODEL;
  float ss = 0.f;
#pragma unroll
  for (int d = 0; d < D_MODEL; ++d) ss += hrow[d] * hrow[d];
  float inv = rsqrtf(ss * (1.0f / D_MODEL) + 1e-5f);
  float acc = 0.f;
#pragma unroll
  for (int d = 0; d < D_MODEL; ++d)
    acc += hrow[d] * inv * norm_w[d] * in_w[d * (2 * ED) + j];
  if (j < ED) xs_raw[tok * ED + j] = acc;
  else        zbuf[tok * ED + (j - ED)] = acc;
}

// ---------------------------------------------------------------------------
// Kernel 2: causal depthwise conv (DCONV=16) + bias + silu.
// ---------------------------------------------------------------------------
__global__ void conv_silu_k(const float* __restrict__ xs_raw,
                            const float* __restrict__ conv_w,
                            const float* __restrict__ conv_b,
                            float* __restrict__ xs) {
  int t = blockIdx.x * blockDim.x + threadIdx.x;
  if (t >= BL * ED) return;
  int tok = t / ED;
  int e   = t % ED;
  int b = tok / SEQL, l = tok % SEQL;
  float acc = conv_b[e];
#pragma unroll
  for (int k = 0; k < DCONV; ++k) {
    int ls = l - (DCONV - 1) + k;
    if (ls >= 0) acc += xs_raw[(b * SEQL + ls) * ED + e] * conv_w[e * DCONV + k];
  }
  xs[tok * ED + e] = acc / (1.f + __expf(-acc));   // silu
}

// ---------------------------------------------------------------------------
// Kernel 3a: pack+pad x_proj_w (48x65) into pair-interleaved padded layout:
//   Wpp[(k>>1)*160 + col*2 + (k&1)] ; cols 65..79 zero.
// A B-fragment for one column (K rows k0,k0+1) is then 8 contiguous bytes.
// ---------------------------------------------------------------------------
__global__ void packw_k(const float* __restrict__ W, float* __restrict__ Wpp) {
  int t = blockIdx.x * blockDim.x + threadIdx.x;
  if (t >= WPP_N) return;
  int p   = t / 160;        // K pair index 0..23
  int r   = t % 160;
  int col = r >> 1;         // 0..79
  int k   = p * 2 + (r & 1);
  Wpp[t] = (col < (1 + 2 * NSTATE)) ? W[k * 65 + col] : 0.f;
}

// ---------------------------------------------------------------------------
// Kernel 3b: x_proj GEMM (16384x48)@(48x80) with V_WMMA_F32_16X16X4_F32.
// Packed weight block (15360 B) is staged into LDS once per workgroup via the
// Tensor Data Mover (async tensor DMA, TENSORcnt), then each of the 8 waves
// computes one 16x16 tile per block with branch-free b64/ds_load_b64 fragment
// loads.  1024 M-tiles x 5 N-tiles = 5120 full waves, EXEC all-ones.
// ---------------------------------------------------------------------------
__global__ void xproj_wmma_k(const float* __restrict__ xs,
                             const float* __restrict__ Wpp,  // packed (24x160)
                             float* __restrict__ dbc) {      // (BL x 80)
  __shared__ float lds_w[WPP_N];                             // 15360 bytes

#if __has_builtin(__builtin_amdgcn_tensor_load_to_lds)
  if (threadIdx.x < 32) {   // wave 0 issues the TDM DMA (per-wave instruction)
    unsigned lds_off = (unsigned)(unsigned long long)(uintptr_t)&lds_w[0];
    unsigned long long ga = (unsigned long long)(uintptr_t)Wpp;
    // D# group 0: count=1, lds_addr, global_addr[56:0], type=2 ("image")
    v4u g0;
    g0.x = 1u;                                  // count=1, is_restore=0
    g0.y = lds_off;                             // lds_addr
    g0.z = (unsigned)(ga & 0xFFFFFFFFu);        // global_addr[31:0]
    g0.w = (unsigned)((ga >> 32) & 0x01FFFFFFu) | 0x80000000u;  // [56:32] | type=2
    // D# group 1: data_size=4B, 1-D tensor/tile of 3840 dwords, mask=0 (no cluster)
    v8i g1;
    g1[0] = (int)(2u << 16);       // data_size=2 (4 bytes); workgroup_mask=0
    g1[1] = (int)(3840u << 16);    // tensor_dim0[15:0]
    g1[2] = (int)(1u << 16);       // tensor_dim0[31:16]=0 | tensor_dim1[15:0]=1
    g1[3] = (int)(3840u << 16);    // tensor_dim1[31:16]=0 | tile_dim0=3840
    g1[4] = 0;                     // tile_dim1=0 (unused), tile_dim2=0
    g1[5] = 3840;                  // tensor_dim0_stride[31:0]
    g1[6] = 0;                     // stride hi | tensor_dim1_stride lo
    g1[7] = 0;
    v4i g2 = {0, 0, 0, 0};
    v4i g3 = {0, 0, 0, 0};
#if defined(__clang_major__) && (__clang_major__ >= 23)
    v8i g4 = {0, 0, 0, 0, 0, 0, 0, 0};
    __builtin_amdgcn_tensor_load_to_lds(g0, g1, g2, g3, g4, 0);
#else
    __builtin_amdgcn_tensor_load_to_lds(g0, g1, g2, g3, 0);
#endif
    __builtin_amdgcn_s_wait_tensorcnt(0);
  }
  __syncthreads();
#else
  for (int i = threadIdx.x; i < WPP_N; i += blockDim.x) lds_w[i] = Wpp[i];
  __syncthreads();
#endif

  int tile = blockIdx.x * 8 + (threadIdx.x >> 5);  // 0..5119
  int lane = threadIdx.x & 31;
  int mt = tile / 5;            // M tile 0..1023
  int nt = tile % 5;            // N tile 0..4
  int lm = lane & 15;           // A row / B,C column within tile
  int hi = lane >> 4;           // half-wave: selects K pair {0,1} vs {2,3}
  int col = nt * 16 + lm;
  const float* arow = xs + (mt * 16 + lm) * ED;
  v8f c = {};
#pragma unroll
  for (int s = 0; s < ED / 4; ++s) {               // 12 K-steps of K=4
    v2f a  = *(const v2f*)(arow + 4 * s + 2 * hi);            // global b64
    v2f bf = *(const v2f*)(lds_w + (2 * s + hi) * 160 + col * 2); // ds_load_b64
    c = __builtin_amdgcn_wmma_f32_16x16x4_f32(false, a, false, bf,
                                              (short)0, c, false, false);
  }
  float* out = dbc + (size_t)(mt * 16) * DBC_PAD + nt * 16;
#pragma unroll
  for (int r = 0; r < 8; ++r)
    out[(r + 8 * hi) * DBC_PAD + lm] = c[r];
}

// ---------------------------------------------------------------------------
// Kernel 4: delta = softplus(dlt * dt_w + dt_b).
// ---------------------------------------------------------------------------
__global__ void delta_k(const float* __restrict__ dbc,
                        const float* __restrict__ dt_w,
                        const float* __restrict__ dt_b,
                        float* __restrict__ delta) {
  int t = blockIdx.x * blockDim.x + threadIdx.x;
  if (t >= BL * ED) return;
  int tok = t / ED;
  int e   = t % ED;
  float dlt = dbc[(size_t)tok * DBC_PAD + 0];
  float v = dlt * dt_w[e] + dt_b[e];
  delta[tok * ED + e] = (v > 20.f) ? v : log1pf(expf(v));
}

// ---------------------------------------------------------------------------
// Kernel 5: selective scan fused with D*xs skip and silu(z) gate.
// One wave32 per (b,e), lane = state index n (N == 32 == wave width).
// Per 32-step block: delta/xs/z pre-loaded coalesced (one load per lane per
// 32 steps) and broadcast per step with __shfl; reduced y kept in-register
// via cndmask; one store per lane per block.  No EXEC branches in the loop.
// ---------------------------------------------------------------------------
__global__ void scan_k(const float* __restrict__ delta,
                       const float* __restrict__ dbc,
                       const float* __restrict__ xs,
                       const float* __restrict__ zbuf,
                       const float* __restrict__ A_log,
                       const float* __restrict__ Dw,
                       float* __restrict__ y) {
  int wid  = (blockIdx.x * blockDim.x + threadIdx.x) >> 5;
  int lane = threadIdx.x & 31;
  if (wid >= BATCH * ED) return;
  int b = wid / ED, e = wid % ED;
  float a_n = -__expf(A_log[e * NSTATE + lane]);   // A = -exp(A_log)
  float Dv  = Dw[e];
  float h   = 0.f;
  const float* bcptr = dbc + (size_t)(b * SEQL) * DBC_PAD;
  for (int l0 = 0; l0 < SEQL; l0 += 32) {
    int tok = b * SEQL + l0 + lane;
    float dvec = delta[(size_t)tok * ED + e];   // coalesced across lanes
    float xvec = xs[(size_t)tok * ED + e];
    float zvec = zbuf[(size_t)tok * ED + e];
    if (l0 + 32 < SEQL)
      __builtin_prefetch(bcptr + (size_t)(l0 + 32) * DBC_PAD + lane, 0, 0);
    float yreg = 0.f;
#pragma unroll
    for (int j = 0; j < 32; ++j) {
      int l = l0 + j;
      float dv = __shfl(dvec, j, 32);           // broadcast step scalars
      float xv = __shfl(xvec, j, 32);
      float Bn = bcptr[(size_t)l * DBC_PAD + 1 + lane];
      float Cn = bcptr[(size_t)l * DBC_PAD + 1 + NSTATE + lane];
      float dA = __expf(dv * a_n);
      h = fmaf(dA, h, dv * Bn * xv);
      float p = h * Cn;
#pragma unroll
      for (int m = 16; m >= 1; m >>= 1) p += __shfl_xor(p, m, 32);
      yreg = (j == lane) ? p : yreg;            // keep own token's sum
    }
    float sz = zvec / (1.f + __expf(-zvec));
    y[(size_t)tok * ED + e] = (yreg + Dv * xvec) * sz;
  }
}

// ---------------------------------------------------------------------------
// Kernel 6: out_proj + residual (layer 0 only).
// ---------------------------------------------------------------------------
__global__ void outproj_k(const float* __restrict__ y,
                          const float* __restrict__ W,
                          const float* __restrict__ res,
                          float* __restrict__ hout) {
  int t = blockIdx.x * blockDim.x + threadIdx.x;
  if (t >= BL * D_MODEL) return;
  int tok = t / D_MODEL;
  int d   = t % D_MODEL;
  float acc = res[tok * D_MODEL + d];
  const float* yrow = y + (size_t)tok * ED;
#pragma unroll
  for (int e = 0; e < ED; ++e) acc += yrow[e] * W[e * D_MODEL + d];
  hout[tok * D_MODEL + d] = acc;
}

// ---------------------------------------------------------------------------
// Kernel 7: head.  feat = relu(h_last @ fc + b); cls/mass from feat.
// ---------------------------------------------------------------------------
__global__ void head_k(const float* __restrict__ y,
                       const float* __restrict__ fc_w, const float* __restrict__ fc_b,
                       const float* __restrict__ cls_w, const float* __restrict__ cls_b,
                       const float* __restrict__ reg_w, const float* __restrict__ reg_b,
                       float* __restrict__ out) {
  int b = blockIdx.x * blockDim.x + threadIdx.x;
  if (b >= BATCH) return;
  const float* hrow = y + ((size_t)b * SEQL + (SEQL - 1)) * ED;
  float feat[ED];
#pragma unroll 4
  for (int j = 0; j < ED; ++j) {
    float a = fc_b[j];
    for (int e = 0; e < ED; ++e) a += hrow[e] * fc_w[e * ED + j];
    feat[j] = fmaxf(a, 0.f);
  }
#pragma unroll
  for (int c = 0; c < NCLS; ++c) {
    float a = cls_b[c];
    for (int e = 0; e < ED; ++e) a += feat[e] * cls_w[e * NCLS + c];
    out[b * NCLS + c] = a;
  }
  float m = reg_b[0];
  for (int e = 0; e < ED; ++e) m += feat[e] * reg_w[e];
  out[BATCH * NCLS + b] = m;
}

// ---------------------------------------------------------------------------
extern "C" void kernel_launch(void* const* d_in, const int* in_sizes, int n_in,
                              void* d_out, int out_size, void* d_ws, size_t ws_size,
                              hipStream_t stream) {
  (void)in_sizes; (void)n_in; (void)out_size; (void)ws_size;
  const float* x         = (const float*)d_in[0];   // (16,1024,6)
  const float* in_proj_w = (const float*)d_in[1];   // (2,6,96)
  const float* conv_w    = (const float*)d_in[2];   // (2,48,16)
  const float* conv_b    = (const float*)d_in[3];   // (2,48)
  const float* x_proj_w  = (const float*)d_in[4];   // (2,48,65)
  const float* dt_w      = (const float*)d_in[5];   // (2,1,48)
  const float* dt_b      = (const float*)d_in[6];   // (2,48)
  const float* A_log     = (const float*)d_in[7];   // (2,48,32)
  const float* Dw        = (const float*)d_in[8];   // (2,48)
  const float* norm_w    = (const float*)d_in[9];   // (2,6)
  const float* out_proj  = (const float*)d_in[10];  // (1,48,6)
  const float* fc_w      = (const float*)d_in[11];
  const float* fc_b      = (const float*)d_in[12];
  const float* cls_w     = (const float*)d_in[13];
  const float* cls_b     = (const float*)d_in[14];
  const float* reg_w     = (const float*)d_in[15];
  const float* reg_b     = (const float*)d_in[16];
  float* out = (float*)d_out;

  // workspace layout (floats)
  float* ws     = (float*)d_ws;
  float* xs_raw = ws;                               // BL*48
  float* zbuf   = ws + (size_t)BL * ED;             // BL*48
  float* xsb    = ws + (size_t)2 * BL * ED;         // BL*48
  float* dbc    = ws + (size_t)3 * BL * ED;         // BL*80
  float* delta  = dbc + (size_t)BL * DBC_PAD;       // BL*48
  float* ybuf   = delta + (size_t)BL * ED;          // BL*48
  float* hbuf   = ybuf + (size_t)BL * ED;           // BL*6
  float* wpp    = hbuf + (size_t)BL * D_MODEL;      // 3840 packed weights

  for (int layer = 0; layer < NLAYERS; ++layer) {
    const float* hin = (layer == 0) ? x : hbuf;
    token_pre_k<<<(BL * 2 * ED + 255) / 256, 256, 0, stream>>>(
        hin, norm_w + layer * D_MODEL, in_proj_w + layer * D_MODEL * 2 * ED,
        xs_raw, zbuf);
    conv_silu_k<<<(BL * ED + 255) / 256, 256, 0, stream>>>(
        xs_raw, conv_w + layer * ED * DCONV, conv_b + layer * ED, xsb);
    packw_k<<<(WPP_N + 255) / 256, 256, 0, stream>>>(
        x_proj_w + layer * ED * 65, wpp);
    xproj_wmma_k<<<5120 / 8, 256, 0, stream>>>(xsb, wpp, dbc);
    delta_k<<<(BL * ED + 255) / 256, 256, 0, stream>>>(
        dbc, dt_w + layer * ED, dt_b + layer * ED, delta);
    scan_k<<<(BATCH * ED * 32 + 255) / 256, 256, 0, stream>>>(
        delta, dbc, xsb, zbuf, A_log + layer * ED * NSTATE, Dw + layer * ED, ybuf);
    if (layer == 0)
      outproj_k<<<(BL * D_MODEL + 255) / 256, 256, 0, stream>>>(
          ybuf, out_proj, x, hbuf);
  }
  head_k<<<1, 64, 0, stream>>>(ybuf, fc_w, fc_b, cls_w, cls_b, reg_w, reg_b, out);
}